// ActorGCN_67748814127825
// MI455X (gfx1250) — compile-verified
//
#include <hip/hip_runtime.h>
#include <stdint.h>

typedef __attribute__((ext_vector_type(2))) float v2f;
typedef __attribute__((ext_vector_type(8))) float v8f;

#define F_DIM 20

// ---------------------------------------------------------------- init
// deg[i] = 1.0 (self loop); zero M2(32x32=1024) + colsum(32) stats area.
__global__ void k_init(float* __restrict__ deg, float* __restrict__ stats, int N) {
  int i = blockIdx.x * blockDim.x + threadIdx.x;
  if (i < N) deg[i] = 1.0f;
  if (i < 1056) stats[i] = 0.0f;
}

// ---------------------------------------------------------------- degree
__global__ void k_deg(const long long* __restrict__ ei, float* __restrict__ deg, int E) {
  int e = blockIdx.x * blockDim.x + threadIdx.x;
  if (e < E) atomicAdd(&deg[(int)ei[(size_t)E + e]], 1.0f);
}

// ---------------------------------------------------------------- dinv + self-loop seed
// dinv[i] = rsqrt(deg[i]) (in place); agg[i,:] = x[i,:] * dinv^2
__global__ void k_selfloop(const float* __restrict__ state, const float* __restrict__ edge_attr,
                           float* __restrict__ dinv, float* __restrict__ agg, int Ns, int N) {
  int i = blockIdx.x * blockDim.x + threadIdx.x;
  if (i >= N) return;
  float r = rsqrtf(dinv[i]);
  dinv[i] = r;
  float n = r * r;
  const float* x = (i < Ns) ? (state + (size_t)i * F_DIM)
                            : (edge_attr + (size_t)(i - Ns) * F_DIM);
  float* a = agg + (size_t)i * F_DIM;
#pragma unroll
  for (int f = 0; f < F_DIM; ++f) a[f] = x[f] * n;
}

// ---------------------------------------------------------------- edge scatter (L2-resident atomics)
__global__ void k_scatter(const float* __restrict__ state, const float* __restrict__ edge_attr,
                          const long long* __restrict__ ei, const float* __restrict__ dinv,
                          float* __restrict__ agg, int E, int Ns) {
  int e = blockIdx.x * blockDim.x + threadIdx.x;
  if (e >= E) return;
  int s = (int)ei[e];
  int d = (int)ei[(size_t)E + e];
  float n = dinv[s] * dinv[d];
  const float* x = (s < Ns) ? (state + (size_t)s * F_DIM)
                            : (edge_attr + (size_t)(s - Ns) * F_DIM);
  float* a = agg + (size_t)d * F_DIM;
#pragma unroll
  for (int f = 0; f < F_DIM; ++f) atomicAdd(&a[f], x[f] * n);
}

// ---------------------------------------------------------------- stats: M2 = agg^T agg, colsum
// fp32 WMMA 16x16x4: per 4-row chunk, A-tile and B-tile share the same lane data.
__global__ void k_stats(const float* __restrict__ agg, float* __restrict__ gM2,
                        float* __restrict__ gCol, int N) {
  __shared__ float sM2[1024];
  __shared__ float sCol[32];
  int tid = threadIdx.x;
  for (int i = tid; i < 1024; i += blockDim.x) sM2[i] = 0.f;
  if (tid < 32) sCol[tid] = 0.f;
  __syncthreads();

  int lane = tid & 31;
  int c = lane & 15;
  bool hi = lane >= 16;
  int wavesPerBlock = blockDim.x >> 5;
  int wave = blockIdx.x * wavesPerBlock + (tid >> 5);
  int nWaves = gridDim.x * wavesPerBlock;
  int chunks = N >> 2;

#if __has_builtin(__builtin_amdgcn_wmma_f32_16x16x4_f32)
  v8f c00 = {}, c01 = {}, c10 = {}, c11 = {};
  float slo = 0.f, shi = 0.f;
  for (int ch = wave; ch < chunks; ch += nWaves) {
    int r0 = (ch << 2) + (hi ? 2 : 0);  // lanes 0-15: K=0,1; lanes 16-31: K=2,3
    const float* p0 = agg + (size_t)r0 * F_DIM;
    v2f vlo, vhi;
    vlo.x = p0[c];
    vlo.y = p0[F_DIM + c];
    if (c < F_DIM - 16) { vhi.x = p0[c + 16]; vhi.y = p0[F_DIM + c + 16]; }
    else                { vhi.x = 0.f;        vhi.y = 0.f; }
    slo += vlo.x + vlo.y;
    shi += vhi.x + vhi.y;
    c00 = __builtin_amdgcn_wmma_f32_16x16x4_f32(false, vlo, false, vlo, (short)0, c00, false, false);
    c01 = __builtin_amdgcn_wmma_f32_16x16x4_f32(false, vlo, false, vhi, (short)0, c01, false, false);
    c10 = __builtin_amdgcn_wmma_f32_16x16x4_f32(false, vhi, false, vlo, (short)0, c10, false, false);
    c11 = __builtin_amdgcn_wmma_f32_16x16x4_f32(false, vhi, false, vhi, (short)0, c11, false, false);
  }
#pragma unroll
  for (int v = 0; v < 8; ++v) {
    int M = v + (hi ? 8 : 0);  // C layout: VGPR v, lanes 0-15 -> M=v, lanes 16-31 -> M=v+8
    atomicAdd(&sM2[M * 32 + c],             c00[v]);
    atomicAdd(&sM2[M * 32 + c + 16],        c01[v]);
    atomicAdd(&sM2[(M + 16) * 32 + c],      c10[v]);
    atomicAdd(&sM2[(M + 16) * 32 + c + 16], c11[v]);
  }
  atomicAdd(&sCol[c], slo);
  atomicAdd(&sCol[c + 16], shi);
  // tail rows (N not divisible by 4) — none for this problem, handled for safety
  if (wave == 0) {
    for (int r = (chunks << 2); r < N; ++r) {
      float x = (lane < F_DIM) ? agg[(size_t)r * F_DIM + lane] : 0.f;
      for (int f = 0; f < F_DIM; ++f)
        atomicAdd(&sM2[f * 32 + lane], __shfl(x, f, 32) * x);
      atomicAdd(&sCol[lane], x);
    }
  }
#else
  // VALU fallback: shuffle-broadcast outer products
  float acc[F_DIM];
#pragma unroll
  for (int f = 0; f < F_DIM; ++f) acc[f] = 0.f;
  float scol = 0.f;
  for (int r = wave; r < N; r += nWaves) {
    float x = (lane < F_DIM) ? agg[(size_t)r * F_DIM + lane] : 0.f;
    scol += x;
#pragma unroll
    for (int f = 0; f < F_DIM; ++f) acc[f] += __shfl(x, f, 32) * x;
  }
#pragma unroll
  for (int f = 0; f < F_DIM; ++f) atomicAdd(&sM2[f * 32 + lane], acc[f]);
  atomicAdd(&sCol[lane], scol);
#endif
  __syncthreads();
  for (int i = tid; i < 1024; i += blockDim.x) atomicAdd(&gM2[i], sM2[i]);
  if (tid < 32) atomicAdd(&gCol[tid], sCol[tid]);
}

// ---------------------------------------------------------------- fold BN+Linear into 20x2
__global__ void k_fold(const float* __restrict__ W_gcn, const float* __restrict__ b_gcn,
                       const float* __restrict__ gamma, const float* __restrict__ beta,
                       const float* __restrict__ W_lin, const float* __restrict__ b_lin,
                       const float* __restrict__ gM2, const float* __restrict__ gCol,
                       float* __restrict__ gWs, int N, int H) {
  __shared__ float acc[64];
  int j = threadIdx.x;
  if (j < 64) acc[j] = 0.f;
  __syncthreads();
  if (j < H) {
    float invN = 1.0f / (float)N;
    float w[F_DIM], cm[F_DIM];
#pragma unroll
    for (int f = 0; f < F_DIM; ++f) {
      w[f] = W_gcn[(size_t)f * H + j];
      cm[f] = gCol[f] * invN;
    }
    double q = 0.0, mw = 0.0;
    for (int f = 0; f < F_DIM; ++f) {
      double row = 0.0;
      for (int f2 = 0; f2 < F_DIM; ++f2) row += (double)gM2[f * 32 + f2] * (double)w[f2];
      q += (double)w[f] * row;
      mw += (double)w[f] * (double)cm[f];
    }
    float bj = b_gcn[j];
    float mean = (float)mw + bj;                                  // E[h_j]
    float e2 = (float)(q * (double)invN) + 2.0f * bj * (float)mw + bj * bj;  // E[h_j^2]
    float var = e2 - mean * mean;                                 // biased variance
    float g = gamma[j] * rsqrtf(fmaxf(var, 0.f) + 1e-5f);
    float wl0 = W_lin[(size_t)j * 2 + 0];
    float wl1 = W_lin[(size_t)j * 2 + 1];
    float u0 = g * wl0, u1 = g * wl1;
    float cb = beta[j] + (bj - mean) * g;  // (b_gcn*g + beta - mean*g)
#pragma unroll
    for (int f = 0; f < F_DIM; ++f) {
      atomicAdd(&acc[f * 2 + 0], w[f] * u0);
      atomicAdd(&acc[f * 2 + 1], w[f] * u1);
    }
    atomicAdd(&acc[40], cb * wl0);
    atomicAdd(&acc[41], cb * wl1);
  }
  __syncthreads();
  if (j < 42) gWs[j] = acc[j] + ((j >= 40) ? b_lin[j - 40] : 0.f);
}

// ---------------------------------------------------------------- per-node GEMV + ReLU + softmax
__global__ void k_final(const float* __restrict__ agg, const float* __restrict__ gWs,
                        float* __restrict__ out, int N) {
  __shared__ float sW[42];
  if (threadIdx.x < 42) sW[threadIdx.x] = gWs[threadIdx.x];
  __syncthreads();
  int i = blockIdx.x * blockDim.x + threadIdx.x;
  if (i >= N) return;
  const float* a = agg + (size_t)i * F_DIM;
  float l0 = sW[40], l1 = sW[41];
#pragma unroll
  for (int f = 0; f < F_DIM; ++f) {
    float av = a[f];
    l0 = fmaf(av, sW[f * 2 + 0], l0);
    l1 = fmaf(av, sW[f * 2 + 1], l1);
  }
  l0 = fmaxf(l0, 0.f);
  l1 = fmaxf(l1, 0.f);
  float m = fmaxf(l0, l1);
  float e0 = expf(l0 - m), e1 = expf(l1 - m);
  float inv = 1.0f / (e0 + e1);
  out[2 * (size_t)i + 0] = e0 * inv;
  out[2 * (size_t)i + 1] = e1 * inv;
}

// ---------------------------------------------------------------- launch
extern "C" void kernel_launch(void* const* d_in, const int* in_sizes, int n_in,
                              void* d_out, int out_size, void* d_ws, size_t ws_size,
                              hipStream_t stream) {
  const float* state      = (const float*)d_in[0];
  const float* edge_attr  = (const float*)d_in[1];
  const long long* ei     = (const long long*)d_in[2];   // int64 [2, E]
  const float* W_gcn      = (const float*)d_in[3];
  const float* b_gcn      = (const float*)d_in[4];
  const float* gamma      = (const float*)d_in[5];
  const float* beta       = (const float*)d_in[6];
  const float* W_lin      = (const float*)d_in[7];
  const float* b_lin      = (const float*)d_in[8];
  float* out = (float*)d_out;

  int Ns = in_sizes[0] / F_DIM;   // 8192
  int E  = in_sizes[2] / 2;       // 200000
  int N  = Ns + E;                // 208192
  int H  = in_sizes[4];           // 1024

  // workspace layout (floats): deg/dinv[N] | agg[N*20] | M2[1024] | colsum[32] | Wsmall[64]
  float* ws   = (float*)d_ws;
  float* deg  = ws;
  float* agg  = deg + N;
  float* gM2  = agg + (size_t)N * F_DIM;
  float* gCol = gM2 + 1024;
  float* gWs  = gCol + 32;

  const int T = 256;
  k_init    <<<(N + T - 1) / T, T, 0, stream>>>(deg, gM2, N);
  k_deg     <<<(E + T - 1) / T, T, 0, stream>>>(ei, deg, E);
  k_selfloop<<<(N + T - 1) / T, T, 0, stream>>>(state, edge_attr, deg, agg, Ns, N);
  k_scatter <<<(E + T - 1) / T, T, 0, stream>>>(state, edge_attr, ei, deg, agg, E, Ns);
  k_stats   <<<256, 256, 0, stream>>>(agg, gM2, gCol, N);
  k_fold    <<<1, H, 0, stream>>>(W_gcn, b_gcn, gamma, beta, W_lin, b_lin, gM2, gCol, gWs, N, H);
  k_final   <<<(N + T - 1) / T, T, 0, stream>>>(agg, gWs, out, N);
}